// ImgStackEncoder_35450660062095
// MI455X (gfx1250) — compile-verified
//
#include <hip/hip_runtime.h>
#include <hip/hip_bf16.h>

typedef __attribute__((ext_vector_type(2))) float v2f;
typedef __attribute__((ext_vector_type(8))) float v8f;

#define NPTS  (1u<<20)
#define BGRID 2
#define FGRID 4
#define HGRID 64
#define WGRID 512
#define MVOX  (BGRID*FGRID*HGRID*WGRID)   // 262144
#define C0 32
#define C1 64
#define CC 16
#define EPSF 1e-5f

// workspace layout (byte offsets); total ~74 MB
#define OFF_CNT   0u                          // MVOX * 4        (uint counts)
#define OFF_VSUM  (1u<<20)                    // MVOX * 4 * 8    (i64 fixed-point sums)
#define OFF_BNFX  (9u<<20)                    // 64 * 8          (i64 fp: sum[32], sumsq[32])
#define OFF_BNSS  ((9u<<20) + 512u)           // 64 * 4          (float: scale[32], shift[32])
#define OFF_VMAX  (10u<<20)                   // MVOX * 64 * 4   (uint encoded max)
#define WS_BYTES  ((size_t)OFF_VMAX + (size_t)MVOX*64u*4u)

#define FX_PTS 1048576.0f    // 2^20 fixed point for voxel sums
#define FX_BN  16777216.0f   // 2^24 fixed point for batchnorm stats

// ---- monotonic float<->uint encoding: unsigned max == float max, 0 == -inf ----
__device__ __forceinline__ unsigned enc_f32(float f) {
  unsigned u = __float_as_uint(f);
  return (u & 0x80000000u) ? ~u : (u | 0x80000000u);
}
__device__ __forceinline__ float dec_f32(unsigned u) {
  return (u & 0x80000000u) ? __uint_as_float(u ^ 0x80000000u) : __uint_as_float(~u);
}

__device__ __forceinline__ void build_x8(const float* __restrict__ pts,
                                         const int* __restrict__ coors,
                                         const unsigned* __restrict__ cnt,
                                         const unsigned long long* __restrict__ vsum,
                                         int i, float x8[8], int* sid_out) {
  int b = coors[4*i+0], h = coors[4*i+1], w = coors[4*i+2], f = coors[4*i+3];
  int sid = ((b*FGRID + f)*HGRID + h)*WGRID + w;
  float p0 = pts[4*i+0], p1 = pts[4*i+1], p2 = pts[4*i+2], p3 = pts[4*i+3];
  float cf  = (float)cnt[sid];
  float inv = 1.0f / fmaxf(cf, 1.0f);
  float s   = inv * (1.0f / FX_PTS);
  float m0 = (float)(long long)vsum[sid*4+0] * s;
  float m1 = (float)(long long)vsum[sid*4+1] * s;
  float m2 = (float)(long long)vsum[sid*4+2] * s;
  x8[0] = p0; x8[1] = p1; x8[2] = p2; x8[3] = p3;
  x8[4] = sqrtf(p0*p0 + p1*p1 + p2*p2);
  x8[5] = p0 - m0; x8[6] = p1 - m1; x8[7] = p2 - m2;
  *sid_out = sid;
}

// ---------------- kernel 0: zero workspace ----------------
__global__ void k_zero(uint4* __restrict__ p, unsigned n4) {
  unsigned i = blockIdx.x*blockDim.x + threadIdx.x;
  unsigned stride = gridDim.x*blockDim.x;
  uint4 z = {0u,0u,0u,0u};
  for (; i < n4; i += stride) p[i] = z;
}

// ---------------- kernel 1: voxel count + fixed-point sums ----------------
__global__ void k_scatter(const float* __restrict__ pts, const int* __restrict__ coors,
                          unsigned* __restrict__ cnt, unsigned long long* __restrict__ vsum) {
  int i = blockIdx.x*blockDim.x + threadIdx.x;
  int b = coors[4*i+0], h = coors[4*i+1], w = coors[4*i+2], f = coors[4*i+3];
  int sid = ((b*FGRID + f)*HGRID + h)*WGRID + w;
  atomicAdd(&cnt[sid], 1u);
  #pragma unroll
  for (int j = 0; j < 4; j++) {
    long long q = (long long)llrintf(pts[4*i+j] * FX_PTS);
    atomicAdd(&vsum[sid*4+j], (unsigned long long)q);
  }
}

// ---------------- kernel 2: batchnorm statistics over x1 = x8 @ w0 ----------------
__global__ void __launch_bounds__(256) k_bnstats(const float* __restrict__ pts,
                                                 const int* __restrict__ coors,
                                                 const float* __restrict__ w0,
                                                 const unsigned* __restrict__ cnt,
                                                 const unsigned long long* __restrict__ vsum,
                                                 unsigned long long* __restrict__ bnfx) {
  __shared__ float sw0[8*C0];
  __shared__ unsigned long long ssum[C0];
  __shared__ unsigned long long ssq[C0];
  int tid = threadIdx.x;
  if (tid < 8*C0) sw0[tid] = w0[tid];
  if (tid < C0) { ssum[tid] = 0ull; ssq[tid] = 0ull; }
  __syncthreads();

  int i = blockIdx.x*blockDim.x + tid;
  float x8[8]; int sid;
  build_x8(pts, coors, cnt, vsum, i, x8, &sid);
  #pragma unroll 4
  for (int c = 0; c < C0; c++) {
    float v = 0.0f;
    #pragma unroll
    for (int r = 0; r < 8; r++) v += x8[r] * sw0[r*C0 + c];
    atomicAdd(&ssum[c], (unsigned long long)(long long)llrintf(v * FX_BN));
    atomicAdd(&ssq[c],  (unsigned long long)(long long)llrintf(v*v * FX_BN));
  }
  __syncthreads();
  if (tid < C0) {
    atomicAdd(&bnfx[tid],      ssum[tid]);
    atomicAdd(&bnfx[C0 + tid], ssq[tid]);
  }
}

// ---------------- kernel 3: fold BN into per-column scale/shift ----------------
__global__ void k_bnfinal(const unsigned long long* __restrict__ bnfx,
                          const float* __restrict__ g0, const float* __restrict__ b0,
                          float* __restrict__ bnss) {
  int c = threadIdx.x;
  if (c < C0) {
    double s  = (double)(long long)bnfx[c]      / (double)FX_BN;
    double sq = (double)(long long)bnfx[C0 + c] / (double)FX_BN;
    double mu  = s / (double)NPTS;
    double var = sq / (double)NPTS - mu*mu;
    if (var < 0.0) var = 0.0;
    float scale = g0[c] * rsqrtf((float)var + EPSF);
    float shift = b0[c] - (float)mu * scale;
    bnss[c]      = scale;
    bnss[C0 + c] = shift;
  }
}

// ---------------- kernel 4: x1(BN,ReLU) @ w1 + b1 via WMMA, scatter-max ----------------
__global__ void __launch_bounds__(256) k_stage2(const float* __restrict__ pts,
                                                const int* __restrict__ coors,
                                                const float* __restrict__ w0,
                                                const float* __restrict__ w1,
                                                const float* __restrict__ b1,
                                                const float* __restrict__ bnss,
                                                const unsigned* __restrict__ cnt,
                                                const unsigned long long* __restrict__ vsum,
                                                unsigned* __restrict__ vmax) {
  __shared__ float sw1[C0*C1];        // 8 KB, shared by all 8 waves
  __shared__ float sx1[8][16][C0];    // 16 KB: per-wave 16x32 activation tile
  __shared__ int   ssid[8][16];
  int tid = threadIdx.x;
  for (int i = tid; i < C0*C1; i += 256) sw1[i] = w1[i];

  int wv = tid >> 5, lane = tid & 31;
  int row = lane & 15;
  int pt  = (blockIdx.x*8 + wv)*16 + row;

  float x8[8]; int sid;
  build_x8(pts, coors, cnt, vsum, pt, x8, &sid);
  int cbase = (lane < 16) ? 0 : 16;     // lanes 0-15: cols 0-15, lanes 16-31: cols 16-31
  #pragma unroll 4
  for (int c = 0; c < 16; c++) {
    int col = cbase + c;
    float v = 0.0f;
    #pragma unroll
    for (int r = 0; r < 8; r++) v += x8[r] * w0[r*C0 + col];
    v = fmaxf(v * bnss[col] + bnss[C0 + col], 0.0f);   // batchnorm + ReLU
    sx1[wv][row][col] = v;
  }
  if (lane < 16) ssid[wv][row] = sid;
  __syncthreads();

  // 16x64 = (16x32) @ (32x64): 4 N-tiles x 8 K-steps of v_wmma_f32_16x16x4_f32
  #pragma unroll
  for (int t = 0; t < 4; t++) {
    v8f c;
    float bias = b1[16*t + row];
    #pragma unroll
    for (int r = 0; r < 8; r++) c[r] = bias;
    #pragma unroll
    for (int k = 0; k < 8; k++) {
      int kb = 4*k + ((lane < 16) ? 0 : 2);
      v2f a, b;
      a.x = sx1[wv][row][kb];   a.y = sx1[wv][row][kb+1];
      b.x = sw1[kb*C1 + 16*t + row];
      b.y = sw1[(kb+1)*C1 + 16*t + row];
      c = __builtin_amdgcn_wmma_f32_16x16x4_f32(false, a, false, b, (short)0, c, false, false);
    }
    int rb = (lane < 16) ? 0 : 8;
    #pragma unroll
    for (int r = 0; r < 8; r++) {
      int m = ssid[wv][rb + r];
      atomicMax(&vmax[(unsigned)m*64u + 16u*t + (unsigned)row], enc_f32(c[r]));
    }
  }
}

// ---------------- kernel 5: voxel GEMM (M x 64) @ (64 x 16) + ReLU + mask ----------------
__global__ void __launch_bounds__(256) k_voxel(const unsigned* __restrict__ vmax,
                                               const unsigned* __restrict__ cnt,
                                               const float* __restrict__ wc,
                                               const float* __restrict__ bc,
                                               float* __restrict__ out) {
  __shared__ float swc[C1*CC];   // 4 KB
  int tid = threadIdx.x;
  for (int i = tid; i < C1*CC; i += 256) swc[i] = wc[i];
  __syncthreads();

  int wv = tid >> 5, lane = tid & 31;
  int row = lane & 15;
  int vbase = (blockIdx.x*8 + wv)*16;
  int arow = vbase + row;
  float amask = (cnt[arow] > 0u) ? 1.0f : 0.0f;  // reference zeroes vmax pre-GEMM

  v8f c;
  float bias = bc[row];
  #pragma unroll
  for (int r = 0; r < 8; r++) c[r] = bias;

  #pragma unroll
  for (int k = 0; k < 16; k++) {
    int kb = 4*k + ((lane < 16) ? 0 : 2);
    v2f a, b;
    a.x = amask * dec_f32(vmax[(unsigned)arow*64u + kb]);
    a.y = amask * dec_f32(vmax[(unsigned)arow*64u + kb + 1]);
    b.x = swc[kb*CC + row];
    b.y = swc[(kb+1)*CC + row];
    c = __builtin_amdgcn_wmma_f32_16x16x4_f32(false, a, false, b, (short)0, c, false, false);
  }

  int rb = (lane < 16) ? 0 : 8;
  #pragma unroll
  for (int r = 0; r < 8; r++) {
    int m = vbase + rb + r;
    float v = (cnt[m] > 0u) ? fmaxf(c[r], 0.0f) : 0.0f;
    out[(unsigned)m*CC + (unsigned)row] = v;
  }
}

extern "C" void kernel_launch(void* const* d_in, const int* in_sizes, int n_in,
                              void* d_out, int out_size, void* d_ws, size_t ws_size,
                              hipStream_t stream) {
  const float* pts   = (const float*)d_in[0];
  const int*   coors = (const int*)  d_in[1];
  const float* w0    = (const float*)d_in[2];
  const float* g0    = (const float*)d_in[3];
  const float* b0    = (const float*)d_in[4];
  const float* w1    = (const float*)d_in[5];
  const float* b1    = (const float*)d_in[6];
  const float* wc    = (const float*)d_in[7];
  const float* bc    = (const float*)d_in[8];

  char* ws = (char*)d_ws;
  unsigned*           cnt  = (unsigned*)          (ws + OFF_CNT);
  unsigned long long* vsum = (unsigned long long*)(ws + OFF_VSUM);
  unsigned long long* bnfx = (unsigned long long*)(ws + OFF_BNFX);
  float*              bnss = (float*)             (ws + OFF_BNSS);
  unsigned*           vmax = (unsigned*)          (ws + OFF_VMAX);
  float* out = (float*)d_out;

  k_zero   <<<2048, 256, 0, stream>>>((uint4*)ws, (unsigned)(WS_BYTES/16));
  k_scatter<<<NPTS/256, 256, 0, stream>>>(pts, coors, cnt, vsum);
  k_bnstats<<<NPTS/256, 256, 0, stream>>>(pts, coors, w0, cnt, vsum, bnfx);
  k_bnfinal<<<1, 32, 0, stream>>>(bnfx, g0, b0, bnss);
  k_stage2 <<<NPTS/128, 256, 0, stream>>>(pts, coors, w0, w1, b1, bnss, cnt, vsum, vmax);
  k_voxel  <<<MVOX/128, 256, 0, stream>>>(vmax, cnt, wc, bc, out);
}